// Inception_Temporal_Layer_32590211842372
// MI455X (gfx1250) — compile-verified
//
#include <hip/hip_runtime.h>

// CDNA5 / gfx1250: fp32 WMMA, 16x16 output tile, K accumulated in steps of 4.
typedef float v2f __attribute__((ext_vector_type(2)));
typedef float v8f __attribute__((ext_vector_type(8)));

__device__ __forceinline__ v8f wmma_f32_k4(v2f a, v2f b, v8f c) {
    // D = A(16x4) * B(4x16) + C, fp32. NEG/NEG_HI zero, no reuse hints.
    return __builtin_amdgcn_wmma_f32_16x16x4_f32(false, a, false, b, (short)0, c,
                                                 false, false);
}

__device__ __forceinline__ float leaky(float v) { return fmaxf(v, 0.01f * v); }

// One wave (32 threads) per sequence n in [0, B*S*F).
// x[n]   : 512 timesteps x 4 channels, contiguous (2048 floats)
// out[n] : [3 scales][48 timesteps][16 channels] (2304 floats)
__global__ __launch_bounds__(32)
void itl_wmma_kernel(const float* __restrict__ x,
                     const float* __restrict__ w1, const float* __restrict__ b1,
                     const float* __restrict__ w2, const float* __restrict__ b2,
                     const float* __restrict__ w3, const float* __restrict__ b3,
                     float* __restrict__ out)
{
    __shared__ float xs[512 * 4];       // [tau][cin]
    __shared__ float o1s[160 * 16];     // [t][h]   (need t <= 149)
    __shared__ float o2s[144 * 16];     // [t][h]   (need t <= 142)

    const int n    = blockIdx.x;
    const int lane = threadIdx.x;       // 0..31 (wave32)
    const int half = lane >> 4;         // selects K pair {0,1} vs {2,3} / D rows +8
    const int m    = lane & 15;         // A row (time) == B col (cout) == D col

    // ---- stage x[n] into LDS: 512 float4, coalesced ----
    {
        const float4* xg = reinterpret_cast<const float4*>(x + (size_t)n * 2048);
        float4*       xl = reinterpret_cast<float4*>(xs);
#pragma unroll
        for (int i = 0; i < 16; ++i)
            xl[lane + 32 * i] = xg[lane + 32 * i];
    }
    __syncthreads();

    // ================= stage 1: conv(K=6, stride=3), K-dim = Cin*6 = 24 ========
    // kk = c*6 + k  ->  B[kk][h] = w1[h*24 + kk] (flat), A[t][kk] = x[3t+k, c]
    {
        const float bias = b1[m];
        for (int tile = 0; tile < 10; ++tile) {          // t = 0..159
            const int t0 = tile * 16;
            v8f acc = {bias, bias, bias, bias, bias, bias, bias, bias};
#pragma unroll
            for (int kk0 = 0; kk0 < 24; kk0 += 4) {
                const int r0 = kk0 + 2 * half, r1 = r0 + 1;
                v2f a, b;
                a.x = xs[(3 * (t0 + m) + (r0 % 6)) * 4 + (r0 / 6)];
                a.y = xs[(3 * (t0 + m) + (r1 % 6)) * 4 + (r1 / 6)];
                b.x = w1[m * 24 + r0];
                b.y = w1[m * 24 + r1];
                acc = wmma_f32_k4(a, b, acc);
            }
#pragma unroll
            for (int r = 0; r < 8; ++r)
                o1s[(t0 + r + 8 * half) * 16 + m] = leaky(acc[r]);
        }
    }
    __syncthreads();

    // ================= stage 2: conv(K=4, dil=2), K-dim = 16*4 = 64 ============
    // kk = hin*4 + k  ->  B[kk][h] = w2[h*64 + kk], A[t][kk] = o1[t+2k, hin]
    {
        const float bias = b2[m];
        for (int tile = 0; tile < 9; ++tile) {           // t = 0..143
            const int t0 = tile * 16;
            v8f acc0 = {bias, bias, bias, bias, bias, bias, bias, bias};
            v8f acc1 = {0, 0, 0, 0, 0, 0, 0, 0};
#pragma unroll
            for (int kk0 = 0; kk0 < 64; kk0 += 8) {
                {
                    const int r0 = kk0 + 2 * half, r1 = r0 + 1;
                    v2f a, b;
                    a.x = o1s[(t0 + m + 2 * (r0 & 3)) * 16 + (r0 >> 2)];
                    a.y = o1s[(t0 + m + 2 * (r1 & 3)) * 16 + (r1 >> 2)];
                    b.x = w2[m * 64 + r0];
                    b.y = w2[m * 64 + r1];
                    acc0 = wmma_f32_k4(a, b, acc0);
                }
                {
                    const int r0 = kk0 + 4 + 2 * half, r1 = r0 + 1;
                    v2f a, b;
                    a.x = o1s[(t0 + m + 2 * (r0 & 3)) * 16 + (r0 >> 2)];
                    a.y = o1s[(t0 + m + 2 * (r1 & 3)) * 16 + (r1 >> 2)];
                    b.x = w2[m * 64 + r0];
                    b.y = w2[m * 64 + r1];
                    acc1 = wmma_f32_k4(a, b, acc1);
                }
            }
            acc0 = acc0 + acc1;
#pragma unroll
            for (int r = 0; r < 8; ++r)
                o2s[(t0 + r + 8 * half) * 16 + m] = leaky(acc0[r]);
        }
    }
    __syncthreads();

    // ================= stage 3: conv(K=13, dil=4, stride=2), K-dim = 208 =======
    // kk = hin*13 + k  ->  B[kk][h] = w3[h*208 + kk], A[t][kk] = o2[2t+4k, hin]
    {
        const float bias = b3[m];
        float* out3 = out + (size_t)n * 2304 + 2 * 768;  // scale index 2
        for (int tile = 0; tile < 3; ++tile) {           // t = 0..47
            const int t0 = tile * 16;
            v8f acc0 = {bias, bias, bias, bias, bias, bias, bias, bias};
            v8f acc1 = {0, 0, 0, 0, 0, 0, 0, 0};
#pragma unroll 2
            for (int kk0 = 0; kk0 < 208; kk0 += 8) {
                {
                    const int r0 = kk0 + 2 * half, r1 = r0 + 1;
                    const int h0 = r0 / 13, k0 = r0 - 13 * h0;
                    const int h1 = r1 / 13, k1 = r1 - 13 * h1;
                    v2f a, b;
                    a.x = o2s[(2 * (t0 + m) + 4 * k0) * 16 + h0];
                    a.y = o2s[(2 * (t0 + m) + 4 * k1) * 16 + h1];
                    b.x = w3[m * 208 + r0];
                    b.y = w3[m * 208 + r1];
                    acc0 = wmma_f32_k4(a, b, acc0);
                }
                {
                    const int r0 = kk0 + 4 + 2 * half, r1 = r0 + 1;
                    const int h0 = r0 / 13, k0 = r0 - 13 * h0;
                    const int h1 = r1 / 13, k1 = r1 - 13 * h1;
                    v2f a, b;
                    a.x = o2s[(2 * (t0 + m) + 4 * k0) * 16 + h0];
                    a.y = o2s[(2 * (t0 + m) + 4 * k1) * 16 + h1];
                    b.x = w3[m * 208 + r0];
                    b.y = w3[m * 208 + r1];
                    acc1 = wmma_f32_k4(a, b, acc1);
                }
            }
            acc0 = acc0 + acc1;
#pragma unroll
            for (int r = 0; r < 8; ++r)
                out3[(t0 + r + 8 * half) * 16 + m] = leaky(acc0[r]);
        }
    }

    // ---- scales 0 and 1: first 48 timesteps of o1/o2, fully coalesced ----
    {
        float* outn = out + (size_t)n * 2304;
#pragma unroll
        for (int i = 0; i < 24; ++i) {
            const int idx = lane + 32 * i;               // < 768 = 48*16
            outn[idx]       = o1s[idx];                  // [t][h] matches output
            outn[768 + idx] = o2s[idx];
        }
    }
}

extern "C" void kernel_launch(void* const* d_in, const int* in_sizes, int n_in,
                              void* d_out, int out_size, void* d_ws, size_t ws_size,
                              hipStream_t stream) {
    const float* x  = (const float*)d_in[0];
    const float* w1 = (const float*)d_in[1];
    const float* b1 = (const float*)d_in[2];
    const float* w2 = (const float*)d_in[3];
    const float* b2 = (const float*)d_in[4];
    const float* w3 = (const float*)d_in[5];
    const float* b3 = (const float*)d_in[6];
    float* out = (float*)d_out;

    const int nseq = in_sizes[0] / (512 * 4);   // B*S*F = 8192
    itl_wmma_kernel<<<dim3(nseq), dim3(32), 0, stream>>>(x, w1, b1, w2, b2, w3, b3, out);
}